// EdgeConv_35931696398859
// MI455X (gfx1250) — compile-verified
//
#include <hip/hip_runtime.h>
#include <cstdint>
#include <cstddef>

typedef float v2f __attribute__((ext_vector_type(2)));
typedef float v8f __attribute__((ext_vector_type(8)));

#define KNN_K 20
#define FLT_BIG 3.402823466e38f

// ---------------------------------------------------------------------------
// Kernel 1: per-point squared norms  sq[b*4096+n] = sum_d x[b,d,n]^2
// ---------------------------------------------------------------------------
__global__ __launch_bounds__(256) void sq_kernel(const float* __restrict__ x,
                                                 float* __restrict__ sq) {
  int p = blockIdx.x * 256 + threadIdx.x;           // 0..32767
  int b = p >> 12, n = p & 4095;
  const float* xp = x + ((size_t)b * 64) * 4096 + n;
  float s = 0.f;
#pragma unroll
  for (int d = 0; d < 64; ++d) { float v = xp[(size_t)d * 4096]; s += v * v; }
  sq[p] = s;
}

// ---------------------------------------------------------------------------
// Kernel 2: fused gram (WMMA f32 16x16x4) + top-20 KNN per row.
// Block = 256 threads (8 waves). Block handles 64 rows of one batch.
// ---------------------------------------------------------------------------
__global__ __launch_bounds__(256) void knn_kernel(const float* __restrict__ x,
                                                  const float* __restrict__ sq,
                                                  int* __restrict__ knn) {
  __shared__ float s_Xi[64 * 68];   // Xi[m][d], padded stride 68
  __shared__ float s_Xj[64 * 68];   // Xj[n][d]
  __shared__ float s_G[64 * 65];    // gram tile, padded stride 65
  __shared__ float s_hd[64 * KNN_K];
  __shared__ int   s_hi[64 * KNN_K];
  __shared__ float s_sqj[64];

  const int tid = threadIdx.x;
  const int b = blockIdx.x >> 6;
  const int i0 = (blockIdx.x & 63) << 6;
  const float* xb = x + (size_t)b * 64 * 4096;

  for (int q = tid; q < 4096; q += 256) {
    int mm = q & 63, dd = q >> 6;
    s_Xi[mm * 68 + dd] = xb[(size_t)dd * 4096 + i0 + mm];
  }
  if (tid < 64) {
    for (int k = 0; k < KNN_K; ++k) { s_hd[tid * KNN_K + k] = FLT_BIG; s_hi[tid * KNN_K + k] = 0; }
  }
  float thr = FLT_BIG; int pos = 0;   // per-row-owner heap threshold

  const int wv = tid >> 5, lane = tid & 31;
  const int g2 = (lane >> 4) << 1;    // 0 or 2 (K-pair select)
  const int mlane = lane & 15;

  for (int jb = 0; jb < 64; ++jb) {
    const int j0 = jb << 6;
    __syncthreads();                  // prev selection done; safe to overwrite Xj/G
    for (int q = tid; q < 4096; q += 256) {
      int mm = q & 63, dd = q >> 6;
      s_Xj[mm * 68 + dd] = xb[(size_t)dd * 4096 + j0 + mm];
    }
    if (tid < 64) s_sqj[tid] = sq[b * 4096 + j0 + tid];
    __syncthreads();

    // 16 tiles of 16x16; each wave computes 2 tiles, K-loop of 16 WMMAs.
#pragma unroll
    for (int q = 0; q < 2; ++q) {
      int t = wv * 2 + q;
      int ti = t >> 2, tj = t & 3;
      v8f acc = {};
      const float* ai = &s_Xi[(ti * 16 + mlane) * 68 + g2];
      const float* bj = &s_Xj[(tj * 16 + mlane) * 68 + g2];
#pragma unroll
      for (int kk = 0; kk < 64; kk += 4) {
        v2f a, bb;
        a.x = ai[kk];  a.y = ai[kk + 1];
        bb.x = bj[kk]; bb.y = bj[kk + 1];
        acc = __builtin_amdgcn_wmma_f32_16x16x4_f32(false, a, false, bb,
                                                    (short)0, acc, false, false);
      }
      int rb = ti * 16 + ((lane >> 4) << 3);
#pragma unroll
      for (int v = 0; v < 8; ++v)
        s_G[(rb + v) * 65 + tj * 16 + mlane] = acc[v];
    }
    __syncthreads();

    // Top-K maintenance: one owner thread per row; rank by sq[j] - 2*G[i][j]
    // (equal ordering to d2 since sq[i] is row-constant).
    if (tid < 64) {
      const int m = tid;
      for (int n = 0; n < 64; ++n) {
        float r = s_sqj[n] - 2.0f * s_G[m * 65 + n];
        if (r < thr) {
          s_hd[m * KNN_K + pos] = r;
          s_hi[m * KNN_K + pos] = j0 + n;
          float mx = -FLT_BIG; int mp = 0;
          for (int k = 0; k < KNN_K; ++k) {
            float v = s_hd[m * KNN_K + k];
            if (v > mx) { mx = v; mp = k; }
          }
          thr = mx; pos = mp;
        }
      }
    }
  }
  __syncthreads();
  if (tid < 64) {
    int* dst = knn + ((size_t)b * 4096 + i0 + tid) * KNN_K;
    for (int k = 0; k < KNN_K; ++k) dst[k] = s_hi[tid * KNN_K + k];
  }
}

// ---------------------------------------------------------------------------
// Kernel 3: projections pw = X*w1^T, cw = X*(w2-w1)^T  (WMMA f32 16x16x4)
// ---------------------------------------------------------------------------
__global__ __launch_bounds__(256) void proj_kernel(const float* __restrict__ x,
                                                   const float* __restrict__ w,
                                                   float* __restrict__ pw,
                                                   float* __restrict__ cw) {
  __shared__ float s_X[64 * 68];
  __shared__ float s_Wl[64 * 68];   // w1[o][d]
  __shared__ float s_Wd[64 * 68];   // (w2-w1)[o][d]
  const int tid = threadIdx.x;
  const int b = blockIdx.x >> 6;
  const int n0 = (blockIdx.x & 63) << 6;
  const float* xb = x + (size_t)b * 64 * 4096;

  for (int q = tid; q < 4096; q += 256) {
    int mm = q & 63, dd = q >> 6;
    s_X[mm * 68 + dd] = xb[(size_t)dd * 4096 + n0 + mm];
    float wl = w[mm * 128 + dd];
    s_Wl[mm * 68 + dd] = wl;
    s_Wd[mm * 68 + dd] = w[mm * 128 + 64 + dd] - wl;
  }
  __syncthreads();

  const int wv = tid >> 5, lane = tid & 31;
  const int g2 = (lane >> 4) << 1, mlane = lane & 15;
#pragma unroll
  for (int q = 0; q < 4; ++q) {
    int t = wv * 4 + q;           // 0..31 : 16 pw tiles then 16 cw tiles
    int which = t >> 4;
    int tt = t & 15;
    int ti = tt >> 2, tj = tt & 3;
    const float* ws = which ? &s_Wd[0] : &s_Wl[0];
    v8f acc = {};
    const float* ai = &s_X[(ti * 16 + mlane) * 68 + g2];
    const float* bj = &ws[(tj * 16 + mlane) * 68 + g2];
#pragma unroll
    for (int kk = 0; kk < 64; kk += 4) {
      v2f a, bb;
      a.x = ai[kk];  a.y = ai[kk + 1];
      bb.x = bj[kk]; bb.y = bj[kk + 1];
      acc = __builtin_amdgcn_wmma_f32_16x16x4_f32(false, a, false, bb,
                                                  (short)0, acc, false, false);
    }
    float* dst = which ? cw : pw;
    int rb = n0 + ti * 16 + ((lane >> 4) << 3);
#pragma unroll
    for (int v = 0; v < 8; ++v)
      dst[((size_t)b * 4096 + rb + v) * 64 + tj * 16 + mlane] = acc[v];
  }
}

// ---------------------------------------------------------------------------
// Kernel 4: h[b,n,o] = cw[b,n,o] + max_k pw[b, idx[b,n,k], o]
// Block = 4 points x 64 channels.
// ---------------------------------------------------------------------------
__global__ __launch_bounds__(256) void gather_kernel(const float* __restrict__ pw,
                                                     const float* __restrict__ cw,
                                                     const int* __restrict__ knn,
                                                     float* __restrict__ h) {
  __shared__ int s_idx[4 * KNN_K];
  const int tid = threadIdx.x;
  const int p0 = blockIdx.x << 2;
  if (tid < 4 * KNN_K) s_idx[tid] = knn[(size_t)p0 * KNN_K + tid];
  __syncthreads();
  const int pt = tid >> 6, o = tid & 63;
  const int p = p0 + pt;
  const int b = p >> 12;
  const float* pwb = pw + (size_t)b * 4096 * 64 + o;
  float best = -FLT_BIG;
#pragma unroll
  for (int k = 0; k < KNN_K; ++k) {
    int j = s_idx[pt * KNN_K + k];
    best = fmaxf(best, pwb[(size_t)j * 64]);
  }
  h[(size_t)p * 64 + o] = cw[(size_t)p * 64 + o] + best;
}

// ---------------------------------------------------------------------------
// Kernel 5: deterministic per-channel BN stats -> scale/shift
// ---------------------------------------------------------------------------
__global__ __launch_bounds__(256) void stats_kernel(const float* __restrict__ h,
                                                    const float* __restrict__ gamma,
                                                    const float* __restrict__ beta,
                                                    float* __restrict__ scl,
                                                    float* __restrict__ shf) {
  __shared__ float rs[256], rq[256];
  const int o = blockIdx.x, tid = threadIdx.x;
  float s = 0.f, q = 0.f;
  for (int i = 0; i < 128; ++i) {
    float v = h[(size_t)(i * 256 + tid) * 64 + o];
    s += v; q += v * v;
  }
  rs[tid] = s; rq[tid] = q;
  __syncthreads();
  for (int step = 128; step > 0; step >>= 1) {
    if (tid < step) { rs[tid] += rs[tid + step]; rq[tid] += rq[tid + step]; }
    __syncthreads();
  }
  if (tid == 0) {
    float mean = rs[0] * (1.0f / 32768.0f);
    float var  = rq[0] * (1.0f / 32768.0f) - mean * mean;
    var = fmaxf(var, 0.0f);
    float sc = gamma[o] / sqrtf(var + 1e-5f);
    scl[o] = sc;
    shf[o] = beta[o] - mean * sc;
  }
}

// ---------------------------------------------------------------------------
// Kernel 6: BN + exact GELU + transpose write  out[b,o,n]
// ---------------------------------------------------------------------------
__global__ __launch_bounds__(256) void bngelu_kernel(const float* __restrict__ h,
                                                     const float* __restrict__ scl,
                                                     const float* __restrict__ shf,
                                                     float* __restrict__ out) {
  int idx = blockIdx.x * 256 + threadIdx.x;   // output-linear: b,o,n
  int n = idx & 4095;
  int o = (idx >> 12) & 63;
  int b = idx >> 18;
  float v = h[(((size_t)b << 12) + n) * 64 + o];
  float y = v * scl[o] + shf[o];
  out[idx] = 0.5f * y * (1.0f + erff(y * 0.70710678118654752f));
}

// ---------------------------------------------------------------------------
extern "C" void kernel_launch(void* const* d_in, const int* in_sizes, int n_in,
                              void* d_out, int out_size, void* d_ws, size_t ws_size,
                              hipStream_t stream) {
  (void)in_sizes; (void)n_in; (void)out_size; (void)ws_size;
  const float* x     = (const float*)d_in[0];   // (8,64,4096)
  const float* w     = (const float*)d_in[1];   // (64,128)
  const float* gamma = (const float*)d_in[2];   // (64,)
  const float* beta  = (const float*)d_in[3];   // (64,)
  float* out = (float*)d_out;                   // (8,64,4096)

  char* ws = (char*)d_ws;
  size_t off = 0;
  float* sqv  = (float*)(ws + off); off += 32768u * 4;                 // 128 KB
  off = (off + 255) & ~(size_t)255;
  int*   knn  = (int*)(ws + off);   off += (size_t)32768 * KNN_K * 4;  // 2.62 MB
  off = (off + 255) & ~(size_t)255;
  float* pw   = (float*)(ws + off); off += (size_t)32768 * 64 * 4;     // 8 MB
  off = (off + 255) & ~(size_t)255;
  float* cw   = (float*)(ws + off); off += (size_t)32768 * 64 * 4;     // 8 MB
  off = (off + 255) & ~(size_t)255;
  float* hbuf = (float*)(ws + off); off += (size_t)32768 * 64 * 4;     // 8 MB
  off = (off + 255) & ~(size_t)255;
  float* scl  = (float*)(ws + off); off += 256;
  float* shf  = (float*)(ws + off); off += 256;

  sq_kernel   <<<128,  256, 0, stream>>>(x, sqv);
  knn_kernel  <<<512,  256, 0, stream>>>(x, sqv, knn);
  proj_kernel <<<512,  256, 0, stream>>>(x, w, pw, cw);
  gather_kernel<<<8192, 256, 0, stream>>>(pw, cw, knn, hbuf);
  stats_kernel<<<64,   256, 0, stream>>>(hbuf, gamma, beta, scl, shf);
  bngelu_kernel<<<8192, 256, 0, stream>>>(hbuf, scl, shf, out);
}